// TopK_stablized_31258771981054
// MI455X (gfx1250) — compile-verified
//
#include <hip/hip_runtime.h>
#include <math.h>

// ---------------- problem constants ----------------
constexpr int kN    = 4096;
constexpr int kBS   = 256;
constexpr int kK    = 15;
constexpr int kKP1  = 16;
constexpr int kIter = 200;
constexpr int kNT   = 512;            // threads per row-block (16 waves)
constexpr int kJPT  = kN / kNT;       // 8 columns per thread
constexpr int kNW   = kNT / 32;       // 16 waves
#define EPSV 0.1f

// Vector type matching the async-to-LDS builtin's parameter type
// (round-1 diagnostic showed: "__attribute__((__vector_size__(4 * sizeof(int)))) int __device__ *")
typedef int v4i __attribute__((vector_size(4 * sizeof(int))));
typedef __attribute__((address_space(1))) v4i gv4i;   // global
typedef __attribute__((address_space(3))) v4i lv4i;   // LDS

// CDNA5 async global->LDS path (builtin confirmed present by round-1 diagnostics).
#if defined(__HIP_DEVICE_COMPILE__) && __has_builtin(__builtin_amdgcn_global_load_async_to_lds_b128) && __has_builtin(__builtin_amdgcn_s_wait_asynccnt)
#define USE_ASYNC_LDS 1
#else
#define USE_ASYNC_LDS 0
#endif

// ---------------- helpers ----------------
__device__ __forceinline__ float fast_exp2(float x) {
#if defined(__HIP_DEVICE_COMPILE__) && __has_builtin(__builtin_amdgcn_exp2f)
  return __builtin_amdgcn_exp2f(x);       // v_exp_f32
#else
  return exp2f(x);
#endif
}

__device__ __forceinline__ unsigned mapOrd(float f) {
  unsigned u = __float_as_uint(f);
  return (u & 0x80000000u) ? ~u : (u | 0x80000000u);
}
__device__ __forceinline__ float unmapOrd(unsigned m) {
  unsigned u = (m & 0x80000000u) ? (m ^ 0x80000000u) : ~m;
  return __uint_as_float(u);
}
__device__ __forceinline__ float waveMax(float v) {
#pragma unroll
  for (int off = 16; off >= 1; off >>= 1) v = fmaxf(v, __shfl_xor(v, off, 32));
  return v;
}
__device__ __forceinline__ float waveMin(float v) {
#pragma unroll
  for (int off = 16; off >= 1; off >>= 1) v = fminf(v, __shfl_xor(v, off, 32));
  return v;
}
__device__ __forceinline__ float waveSum(float v) {
#pragma unroll
  for (int off = 16; off >= 1; off >>= 1) v += __shfl_xor(v, off, 32);
  return v;
}

// ---------------- main Sinkhorn kernel: one block per batch row ----------------
// (placed first in the file so the device-asm snippet shows its prologue,
//  including the async global->LDS issue sequence)
__global__ __launch_bounds__(kNT) void sinkhorn_kernel(const float* __restrict__ scores,
                                                       float* __restrict__ out,
                                                       const unsigned* __restrict__ ws) {
  const int b = blockIdx.x;
  const int tid = threadIdx.x;
  const int lane = tid & 31;
  const int wave = tid >> 5;

  __shared__ float gS[kKP1];
  __shared__ float Msh[kKP1];
  __shared__ float redA[kKP1][kNW];
  __shared__ float redB[kKP1][kNW];

#if USE_ASYNC_LDS
  __shared__ __align__(16) float sRow[kN];
  // Issue the async DMA first so it overlaps the scalar ws loads below.
  {
    gv4i* gsrc = (gv4i*)(scores + (size_t)b * kN);   // 16B chunks of the score row
    lv4i* ldst = (lv4i*)(&sRow[0]);
    // 4096 floats = 1024 x 16B chunks; 512 threads -> 2 chunks each
#pragma unroll
    for (int q = 0; q < (kN / 4) / kNT; ++q) {
      int c = tid + q * kNT;
      __builtin_amdgcn_global_load_async_to_lds_b128(gsrc + c, ldst + c, 0, 0);
    }
  }
#endif

  const float max_s = unmapOrd(ws[0]);
  const float min_s = unmapOrd(ws[1]);
  const float cmax  = unmapOrd(ws[2]);
  const float filled = min_s - (max_s - min_s);
  const float invC = (cmax > 0.0f) ? (1.0f / cmax) : 0.0f;

  // log2-domain constants: exp(x) = exp2(x*log2e); LSE_nat = ln2*(m + log2(S))
  const float c2        = 1.4426950408889634f / EPSV;        // log2(e)/eps
  const float negEpsLn2 = -EPSV * 0.6931471805599453f;       // -eps*ln2
  const float epsLnN    = EPSV * 8.317766166719343f;         // eps*ln(4096)
  const float epsLogNuLast = EPSV * logf((float)(kN - kK) / (float)kN);

  float s[kJPT], f[kJPT];

#if USE_ASYNC_LDS
  // Each wave waits for its own async loads; the barrier then makes all
  // waves' chunks visible to everyone.
  __builtin_amdgcn_s_wait_asynccnt(0);
  __syncthreads();
#pragma unroll
  for (int j = 0; j < kJPT; ++j) {
    float v = sRow[tid + kNT * j];
    if (__float_as_uint(v) == 0xFF800000u) v = filled;
    s[j] = v;
  }
#else
#pragma unroll
  for (int j = 0; j < kJPT; ++j) {
    float v = scores[(size_t)b * kN + tid + kNT * j];
    if (__float_as_uint(v) == 0xFF800000u) v = filled;
    s[j] = v;
  }
#endif

  if (tid < kKP1) gS[tid] = 0.0f;
  __syncthreads();

#pragma unroll 1
  for (int it = 0; it < kIter; ++it) {
    // ---- f update: f[n] = -eps*LSE_k((g[k]-C[k,n])/eps) - eps*ln(N) ----
    float greg[kKP1];
#pragma unroll
    for (int k = 0; k < kKP1; ++k) greg[k] = gS[k];

#pragma unroll
    for (int j = 0; j < kJPT; ++j) {
      float z[kKP1];
      float m = -INFINITY;
#pragma unroll
      for (int k = 0; k < kKP1; ++k) {
        float a = (float)(kK - k);
        float d = s[j] - a;
        z[k] = (greg[k] - d * d * invC) * c2;
        m = fmaxf(m, z[k]);
      }
      float ssum = 0.0f;
#pragma unroll
      for (int k = 0; k < kKP1; ++k) ssum += fast_exp2(z[k] - m);
      f[j] = negEpsLn2 * (m + __log2f(ssum)) - epsLnN;
    }

    // ---- g update pass A: per-k global max of w = (f[n]-C[k,n])/eps*log2e ----
#pragma unroll
    for (int k = 0; k < kKP1; ++k) {
      float a = (float)(kK - k);
      float lm = -INFINITY;
#pragma unroll
      for (int j = 0; j < kJPT; ++j) {
        float d = s[j] - a;
        float w = (f[j] - d * d * invC) * c2;
        lm = fmaxf(lm, w);
      }
      lm = waveMax(lm);
      if (lane == 0) redA[k][wave] = lm;
    }
    __syncthreads();
    if (tid < kKP1) {
      float m = redA[tid][0];
#pragma unroll
      for (int w2 = 1; w2 < kNW; ++w2) m = fmaxf(m, redA[tid][w2]);
      Msh[tid] = m;
    }
    __syncthreads();

    // ---- g update pass B: per-k sum of exp2(w - M[k]) ----
#pragma unroll
    for (int k = 0; k < kKP1; ++k) {
      float a = (float)(kK - k);
      float Mk = Msh[k];
      float ps = 0.0f;
#pragma unroll
      for (int j = 0; j < kJPT; ++j) {
        float d = s[j] - a;
        float w = (f[j] - d * d * invC) * c2;
        ps += fast_exp2(w - Mk);
      }
      ps = waveSum(ps);
      if (lane == 0) redB[k][wave] = ps;
    }
    __syncthreads();
    if (tid < kKP1) {
      float S = 0.0f;
#pragma unroll
      for (int w2 = 0; w2 < kNW; ++w2) S += redB[tid][w2];
      float lognu = (tid < kK) ? -epsLnN : epsLogNuLast;
      gS[tid] = negEpsLn2 * (Msh[tid] + __log2f(S)) + lognu;
    }
    __syncthreads();
  }

  // ---- output: out[b, n, kk] = exp((f[n]+g[kk]-C[kk,n])/eps) * N, kk in 0..14 ----
  float greg[kKP1];
#pragma unroll
  for (int k = 0; k < kKP1; ++k) greg[k] = gS[k];

#pragma unroll
  for (int j = 0; j < kJPT; ++j) {
    int n = tid + kNT * j;
    size_t base = ((size_t)b * kN + n) * (size_t)kK;
#pragma unroll
    for (int kk = 0; kk < kK; ++kk) {
      float a = (float)(kK - kk);
      float d = s[j] - a;
      float e = (f[j] + greg[kk] - d * d * invC) * c2 + 12.0f;  // *N == +log2(4096)
      out[base + kk] = fast_exp2(e);
    }
  }
}

// ---------------- preprocessing ----------------
__global__ void ws_init_kernel(unsigned* ws) {
  ws[0] = 0u;           // ordered-mapped running max(s)
  ws[1] = 0xFFFFFFFFu;  // ordered-mapped running min(s, -inf -> +inf)
  ws[2] = 0u;           // ordered-mapped running max(C)
}

__global__ void minmax_kernel(const float* __restrict__ s, unsigned* ws, int total) {
  int i0 = blockIdx.x * blockDim.x + threadIdx.x;
  int stride = gridDim.x * blockDim.x;
  float mx = -INFINITY, mn = INFINITY;
  for (int i = i0; i < total; i += stride) {
    float v = s[i];
    mx = fmaxf(mx, v);
    float vn = (__float_as_uint(v) == 0xFF800000u) ? INFINITY : v;  // -inf -> +inf
    mn = fminf(mn, vn);
  }
  mx = waveMax(mx);
  mn = waveMin(mn);
  if ((threadIdx.x & 31) == 0) {
    atomicMax(&ws[0], mapOrd(mx));
    atomicMin(&ws[1], mapOrd(mn));
  }
}

__global__ void cmax_kernel(const float* __restrict__ s, unsigned* ws, int total) {
  float max_s = unmapOrd(ws[0]);
  float min_s = unmapOrd(ws[1]);
  float filled = min_s - (max_s - min_s);
  int i0 = blockIdx.x * blockDim.x + threadIdx.x;
  int stride = gridDim.x * blockDim.x;
  float mx = -INFINITY;
  for (int i = i0; i < total; i += stride) {
    float v = s[i];
    if (__float_as_uint(v) == 0xFF800000u) v = filled;
    // max over anchors {0..15} of (v-a)^2 is attained at an endpoint (convex in a)
    float d1 = v;            // a = 0
    float d2 = v - 15.0f;    // a = 15
    mx = fmaxf(mx, fmaxf(d1 * d1, d2 * d2));
  }
  mx = waveMax(mx);
  if ((threadIdx.x & 31) == 0) atomicMax(&ws[2], mapOrd(mx));
}

// ---------------- launch ----------------
extern "C" void kernel_launch(void* const* d_in, const int* in_sizes, int n_in,
                              void* d_out, int out_size, void* d_ws, size_t ws_size,
                              hipStream_t stream) {
  (void)in_sizes; (void)n_in; (void)out_size; (void)ws_size;
  const float* scores = (const float*)d_in[0];
  float* out = (float*)d_out;
  unsigned* ws = (unsigned*)d_ws;
  const int total = kBS * kN;

  ws_init_kernel<<<1, 1, 0, stream>>>(ws);
  minmax_kernel<<<512, 256, 0, stream>>>(scores, ws, total);
  cmax_kernel<<<512, 256, 0, stream>>>(scores, ws, total);
  sinkhorn_kernel<<<kBS, kNT, 0, stream>>>(scores, out, ws);
}